// _QuantumKernel_65481071406123
// MI455X (gfx1250) — compile-verified
//
#include <hip/hip_runtime.h>
#include <math.h>

typedef __attribute__((ext_vector_type(2))) float v2f;
typedef __attribute__((ext_vector_type(4))) float v4f;
typedef __attribute__((ext_vector_type(8))) float v8f;

// ---------------------------------------------------------------------------
// 4-wire circuit: RX gates fuse per wire; pre-CNOT state is a product state
//   amp(i) = (-i)^popcount(i) * prod_w (bit_w ? sin(a_w/2) : cos(a_w/2))
// and the CNOT ladder is a fixed index permutation (n1=b1^b0, n2=b2^n1,
// n3=b3^n2). Every amplitude is purely real (popcount even) or purely
// imaginary (odd), with the type fixed by the basis index only. Hence each
// product states_x[,k]*states_y[,k] is real, the matmul result is real, and
// with u = signed coefficient (amp = u or i*u):
//   out = | sum_k sigma_k * u_x[k] * u_y[k] |,  sigma_k = +1 real / -1 imag.
// We fold sigma into the Y factor -> ONE real K=16 GEMM, abs = fabsf.
// ---------------------------------------------------------------------------
template <bool FOLD_SIGMA>
__device__ __forceinline__ void encode_u(float a0, float a1, float a2, float a3,
                                         float u[16]) {
  float c[4], s[4];
  sincosf(0.5f * a0, &s[0], &c[0]);
  sincosf(0.5f * a1, &s[1], &c[1]);
  sincosf(0.5f * a2, &s[2], &c[2]);
  sincosf(0.5f * a3, &s[3], &c[3]);
#pragma unroll
  for (int i = 0; i < 16; ++i) {
    const int b0 = (i >> 3) & 1, b1 = (i >> 2) & 1, b2 = (i >> 1) & 1, b3 = i & 1;
    float p = (b0 ? s[0] : c[0]) * (b1 ? s[1] : c[1]) *
              (b2 ? s[2] : c[2]) * (b3 ? s[3] : c[3]);
    const int pc = (b0 + b1 + b2 + b3) & 3;
    // u sign: pc 0,1,2,3 -> +,-,-,+ ; with sigma (+,-,+,-) folded: +,+,-,-
    const bool neg = FOLD_SIGMA ? (pc >= 2) : (pc == 1 || pc == 2);
    const float v = neg ? -p : p;
    const int n1 = b1 ^ b0, n2 = b2 ^ n1, n3 = b3 ^ n2;  // CNOT permutation
    u[(b0 << 3) | (n1 << 2) | (n2 << 1) | n3] = v;
  }
}

// A2[row][16]: fragment-ordered so each lane's K data is contiguous.
// a_idx = kh*8 + kc*2 + e  <-  k = kc*4 + 2*kh + e
__global__ void encode_x_kernel(const float* __restrict__ x, const float* __restrict__ theta,
                                float* __restrict__ A2) {
  const int b = blockIdx.x * blockDim.x + threadIdx.x;
  const float t0 = theta[0], t1 = theta[1], t2 = theta[2], t3 = theta[3];
  float u[16];
  encode_u<false>(x[b * 4 + 0] + t0, x[b * 4 + 1] + t1,
                  x[b * 4 + 2] + t2, x[b * 4 + 3] + t3, u);
  float* row = A2 + (size_t)b * 16;
#pragma unroll
  for (int k = 0; k < 16; ++k) {
    const int a_idx = ((k >> 1) & 1) * 8 + (k >> 2) * 2 + (k & 1);
    row[a_idx] = u[k];
  }
}

// Bp[p][col][2] = { V[2p][col], V[2p+1][col] }, V = sigma-folded Y factors.
// One aligned 64-bit load per B fragment in the GEMM.
__global__ void encode_y_kernel(const float* __restrict__ y, const float* __restrict__ theta,
                                float* __restrict__ Bp, int N) {
  const int b = blockIdx.x * blockDim.x + threadIdx.x;
  const float t0 = theta[0], t1 = theta[1], t2 = theta[2], t3 = theta[3];
  float u[16];
  encode_u<true>(y[b * 4 + 0] + t0, y[b * 4 + 1] + t1,
                 y[b * 4 + 2] + t2, y[b * 4 + 3] + t3, u);
#pragma unroll
  for (int p = 0; p < 8; ++p) {
    v2f pr = {u[2 * p], u[2 * p + 1]};
    *(v2f*)(Bp + ((size_t)p * N + b) * 2) = pr;   // 256B contiguous per wave
  }
}

// ---------------------------------------------------------------------------
// One wave computes a 16 x 64 strip (4 tiles along N): 2 b128 A loads shared,
// 4 b64 B loads + 4 V_WMMA_F32_16X16X4_F32 + 8 fabs + 8 NT stores per tile.
// All waves map to full tiles -> EXEC all ones (WMMA requirement).
// ---------------------------------------------------------------------------
__global__ void wmma_gemm_abs_kernel(const float* __restrict__ A2,
                                     const float* __restrict__ Bp,
                                     float* __restrict__ out,
                                     int N, int tilesN4) {
  const int lane   = threadIdx.x & 31;
  const int waveId = (blockIdx.x * blockDim.x + threadIdx.x) >> 5;
  const int tm = waveId / tilesN4;          // tile row
  const int tj = waveId - tm * tilesN4;     // group of 4 tile cols

  const int lr = lane & 15;                 // M row (A) / N col (B,C) in tile
  const int kh = lane >> 4;                 // K-half select (fragment layout)

  // A fragments: a[kc] = { U[row][kc*4+2kh], U[row][kc*4+2kh+1] }
  const float* abase = A2 + (size_t)(tm * 16 + lr) * 16 + kh * 8;
  v4f fa0 = *(const v4f*)(abase);
  v4f fa1 = *(const v4f*)(abase + 4);
  v2f a[4];
  a[0].x = fa0.x; a[0].y = fa0.y;
  a[1].x = fa0.z; a[1].y = fa0.w;
  a[2].x = fa1.x; a[2].y = fa1.y;
  a[3].x = fa1.z; a[3].y = fa1.w;

#pragma unroll
  for (int t = 0; t < 4; ++t) {
    const int col = (tj * 4 + t) * 16 + lr;
    const float* bbase = Bp + (size_t)col * 2;
    v8f acc = {};
#pragma unroll
    for (int kc = 0; kc < 4; ++kc) {
      // p = kc*2 + kh -> B pair (K = kc*4+2kh, kc*4+2kh+1); immediate offsets
      v2f bfrag = *(const v2f*)(bbase + (size_t)(kc * 2 + kh) * N * 2);
      acc = __builtin_amdgcn_wmma_f32_16x16x4_f32(false, a[kc], false, bfrag,
                                                  (short)0, acc, false, false);
    }
    // C/D layout: VGPR r -> M = r + 8*kh, N = lane%16
    const size_t rowBase = (size_t)(tm * 16 + kh * 8) * N + col;
#pragma unroll
    for (int r = 0; r < 8; ++r) {
      __builtin_nontemporal_store(fabsf(acc[r]), out + rowBase + (size_t)r * N);
    }
  }
}

// ---------------------------------------------------------------------------
extern "C" void kernel_launch(void* const* d_in, const int* in_sizes, int n_in,
                              void* d_out, int out_size, void* d_ws, size_t ws_size,
                              hipStream_t stream) {
  const float* x     = (const float*)d_in[0];
  const float* y     = (const float*)d_in[1];
  const float* theta = (const float*)d_in[2];
  const int Bx = in_sizes[0] / 4;   // 8192
  const int By = in_sizes[1] / 4;   // 8192

  float* A2 = (float*)d_ws;                  // [Bx][16]     = 512 KB
  float* Bp = A2 + (size_t)Bx * 16;          // [8][By][2]   = 512 KB

  encode_x_kernel<<<Bx / 256, 256, 0, stream>>>(x, theta, A2);
  encode_y_kernel<<<By / 256, 256, 0, stream>>>(y, theta, Bp, By);

  const int tilesM = Bx / 16, tilesN4 = By / 64;
  const int waves  = tilesM * tilesN4;       // 65536: one 16x64 strip per wave
  wmma_gemm_abs_kernel<<<waves / 8, 256, 0, stream>>>(A2, Bp, (float*)d_out,
                                                      By, tilesN4);
  (void)n_in; (void)out_size; (void)ws_size;
}